// MemoryAugmentedForecaster_41077067219102
// MI455X (gfx1250) — compile-verified
//
#include <hip/hip_runtime.h>

// ---------------------------------------------------------------------------
// MemoryAugmentedForecaster for MI455X (gfx1250, wave32, WMMA + async-LDS)
//   B=256, S=512, D=512, M=100000, TOPK=16
// ---------------------------------------------------------------------------

#define B_DIM   256
#define S_DIM   512
#define D_DIM   512
#define M_DIM   100000
#define TOPK    16
#define SCALE_F 0.04419417382415922f   // 512^-0.5
#define LN_EPS  1e-5f

#define LDSA_STRIDE 516        // 512 + 4: rows 16B-aligned (2064B), 516%64=4 -> no bank conflicts
#define GX_SIMS 63             // column-blocks over M
#define COLS_PER_BLOCK 1600    // 25 groups x 64 columns
#define CAND_PER_CB 128        // per row, per column-block: 8 threads x 16
#define NCAND (GX_SIMS * CAND_PER_CB)   // 8064 candidates per batch row

typedef float v2f __attribute__((ext_vector_type(2)));
typedef float v8f __attribute__((ext_vector_type(8)));

__device__ __forceinline__ float neg_inf() { return -__builtin_inff(); }

// Async DMA of 16B global -> LDS (gfx1250 ASYNCcnt path).  The LDS byte
// address is the low 32 bits of the generic pointer (ISA aperture mapping).
__device__ __forceinline__ void async_b128_to_lds(const void* gptr, void* lptr) {
    unsigned lds = (unsigned)(uintptr_t)lptr;
    asm volatile("global_load_async_to_lds_b128 %0, %1, off"
                 :: "v"(lds), "v"(gptr) : "memory");
}
__device__ __forceinline__ void wait_asynccnt0() {
    asm volatile("s_wait_asynccnt 0x0" ::: "memory");
}

__device__ __forceinline__ float blockReduceSum256(float v, float* red) {
    const int t = threadIdx.x;
    red[t] = v;
    __syncthreads();
    for (int off = 128; off > 0; off >>= 1) {
        if (t < off) red[t] += red[t + off];
        __syncthreads();
    }
    float r = red[0];
    __syncthreads();
    return r;
}

// ---------------------------------------------------------------------------
// Kernel 1: mean-pool over S + L2 normalize.  One block per batch row.
// ---------------------------------------------------------------------------
__global__ __launch_bounds__(256)
void pool_kernel(const float* __restrict__ hidden, float* __restrict__ series,
                 float* __restrict__ q, int* __restrict__ anyFlag) {
    const int b = blockIdx.x;
    const int t = threadIdx.x;
    if (b == 0 && t == 0) *anyFlag = 0;
    const float2* base2 = (const float2*)(hidden + (size_t)b * S_DIM * D_DIM);
    float acc0 = 0.f, acc1 = 0.f;
#pragma unroll 4
    for (int s = 0; s < S_DIM; ++s) {          // coalesced float2 stream
        const float2 v = base2[(size_t)s * 256 + t];
        acc0 += v.x; acc1 += v.y;
    }
    acc0 *= (1.0f / (float)S_DIM);
    acc1 *= (1.0f / (float)S_DIM);
    const int d0 = 2 * t, d1 = 2 * t + 1;
    series[b * D_DIM + d0] = acc0;
    series[b * D_DIM + d1] = acc1;
    __shared__ float red[256];
    const float sumsq = blockReduceSum256(acc0 * acc0 + acc1 * acc1, red);
    const float invn = rsqrtf(sumsq);          // 1 / ||series||
    q[b * D_DIM + d0] = acc0 * invn;
    q[b * D_DIM + d1] = acc1 * invn;
}

// ---------------------------------------------------------------------------
// Kernel 2: fused sims GEMM (WMMA fp32) + per-block partial top-16.
//   grid = (16 row-blocks [x, fastest -> co-resident], 63 col-blocks [y])
//   block = 128 thr = 4 waves; each wave owns a 16x16 tile per 64-col group.
//   Writes only top-16 candidates per (row, col-block, thread-slot).
// ---------------------------------------------------------------------------
__global__ __launch_bounds__(128)
void sims_topk_partial(const float* __restrict__ q, const float* __restrict__ bank,
                       float* __restrict__ candVal, int* __restrict__ candIdx) {
    __shared__ float ldsA[16 * LDSA_STRIDE];
    __shared__ float sc[16 * 68];              // 16 rows x 64 cols (+4 pad)
    const int tid  = threadIdx.x;
    const int wave = tid >> 5;
    const int lane = tid & 31;
    const int r0    = blockIdx.x * 16;
    const int cbase = blockIdx.y * COLS_PER_BLOCK;

    // Async-DMA the 16x512 q tile into LDS (b128 per lane, 16 issues/thread).
    for (int i = tid; i < 16 * (D_DIM / 4); i += 128) {
        const int m  = i >> 7;                 // / 128 float4 per row
        const int k4 = i & 127;
        async_b128_to_lds(q + (size_t)(r0 + m) * D_DIM + k4 * 4,
                          &ldsA[m * LDSA_STRIDE + k4 * 4]);
    }
    wait_asynccnt0();
    __syncthreads();

    float val[TOPK];
    int   idx[TOPK];
#pragma unroll
    for (int i = 0; i < TOPK; ++i) { val[i] = neg_inf(); idx[i] = 0; }
    const int selRow  = tid >> 3;              // 16 rows x 8 slots
    const int selSlot = tid & 7;

    const int h  = lane >> 4;
    const int mn = lane & 15;
    const float* Alds = &ldsA[mn * LDSA_STRIDE];

    for (int g = 0; g < COLS_PER_BLOCK / 64; ++g) {
        const int c0 = cbase + g * 64;
        if (c0 >= M_DIM) break;                // block-uniform
        const int n0 = c0 + wave * 16;
        if (n0 < M_DIM) {                      // wave-uniform; EXEC all-ones inside
            const float* Brow = bank + (size_t)(n0 + mn) * D_DIM;
            v8f acc = {0.f, 0.f, 0.f, 0.f, 0.f, 0.f, 0.f, 0.f};
#pragma unroll 4
            for (int k0 = 0; k0 < D_DIM; k0 += 4) {
                if ((k0 & 63) == 0)
                    __builtin_prefetch((const void*)(Brow + k0 + 64), 0, 0);
                v2f a, bf;
                a.x  = Alds[k0 + 2 * h];       // A 16x4: lane h*16+m -> A[m][k+2h..]
                a.y  = Alds[k0 + 2 * h + 1];
                bf.x = Brow[k0 + 2 * h];       // B 4x16: lane h*16+n -> B[k+2h..][n]
                bf.y = Brow[k0 + 2 * h + 1];
                acc = __builtin_amdgcn_wmma_f32_16x16x4_f32(
                    false, a, false, bf, (short)0, acc, false, false);
            }
#pragma unroll
            for (int r = 0; r < 8; ++r) {      // C: VGPR r -> rows r and r+8
                float v = acc[r];
                if (v > 0.999f || v < 0.0f) v = neg_inf();  // exclude_self + thresh
                sc[(r + 8 * h) * 68 + wave * 16 + mn] = v;
            }
        }
        __syncthreads();
        // per-thread sorted insert over 8 columns of this thread's row
        for (int j = 0; j < 8; ++j) {
            const int c = selSlot + 8 * j;
            const int gcol = c0 + c;
            if (gcol < M_DIM) {
                const float v = sc[selRow * 68 + c];
                if (v > val[TOPK - 1]) {
                    int p = TOPK - 1;
                    while (p > 0 && v > val[p - 1]) {
                        val[p] = val[p - 1]; idx[p] = idx[p - 1]; --p;
                    }
                    val[p] = v; idx[p] = gcol;
                }
            }
        }
        __syncthreads();                       // before next group's sc writes
    }

    // dump this thread's candidate list
    const size_t base =
        ((size_t)(r0 + selRow) * GX_SIMS + blockIdx.y) * CAND_PER_CB + selSlot * TOPK;
#pragma unroll
    for (int j = 0; j < TOPK; ++j) { candVal[base + j] = val[j]; candIdx[base + j] = idx[j]; }
}

// ---------------------------------------------------------------------------
// Kernel 3: merge 8064 candidates/row -> exact top-16.  One block per row.
// ---------------------------------------------------------------------------
__global__ __launch_bounds__(256)
void topk_merge(const float* __restrict__ candVal, const int* __restrict__ candIdx,
                float* __restrict__ topVal, int* __restrict__ topIdx,
                int* __restrict__ anyFlag) {
    const int b = blockIdx.x;
    const int t = threadIdx.x;
    const size_t base = (size_t)b * NCAND;

    float val[TOPK];
    int   idx[TOPK];
#pragma unroll
    for (int i = 0; i < TOPK; ++i) { val[i] = neg_inf(); idx[i] = 0; }
    for (int i = t; i < NCAND; i += 256) {
        const float v = candVal[base + i];
        if (v > val[TOPK - 1]) {
            const int ix = candIdx[base + i];
            int p = TOPK - 1;
            while (p > 0 && v > val[p - 1]) { val[p] = val[p - 1]; idx[p] = idx[p - 1]; --p; }
            val[p] = v; idx[p] = ix;
        }
    }

    __shared__ float sv[256];
    __shared__ int   si[256];
    __shared__ float winv;
    __shared__ int   winthread;
    int ptr = 0;
    for (int r = 0; r < TOPK; ++r) {
        sv[t] = (ptr < TOPK) ? val[ptr] : neg_inf();
        si[t] = t;
        __syncthreads();
        for (int off = 128; off > 0; off >>= 1) {
            if (t < off && sv[t + off] > sv[t]) { sv[t] = sv[t + off]; si[t] = si[t + off]; }
            __syncthreads();
        }
        if (t == 0) { winv = sv[0]; winthread = si[0]; }
        __syncthreads();
        if (t == winthread) {
            const bool valid = (winv > neg_inf());
            topVal[b * TOPK + r] = winv;
            topIdx[b * TOPK + r] = valid ? idx[ptr] : 0;
            if (valid) atomicOr(anyFlag, 1);
            ++ptr;
        }
        __syncthreads();
    }
}

// ---------------------------------------------------------------------------
// Kernel 4 (generic WMMA fp32): C[r][n] = sum_k A[gather(r)][k]*B[n][k] + bias[n]
// ---------------------------------------------------------------------------
__global__ __launch_bounds__(128)
void wmma_gemm_nt(const float* __restrict__ A, const float* __restrict__ Bm,
                  const int* __restrict__ gatherIdx, const float* __restrict__ bias,
                  float* __restrict__ C, int N) {
    __shared__ float ldsA[16 * LDSA_STRIDE];
    const int r0   = blockIdx.y * 16;
    const int tid  = threadIdx.x;
    const int wave = tid >> 5;
    const int lane = tid & 31;

    for (int i = tid; i < 16 * (D_DIM / 4); i += 128) {
        const int m  = i >> 7;
        const int k4 = i & 127;
        const int row = r0 + m;
        const int src = gatherIdx ? gatherIdx[row] : row;
        async_b128_to_lds(A + (size_t)src * D_DIM + k4 * 4,
                          &ldsA[m * LDSA_STRIDE + k4 * 4]);
    }
    wait_asynccnt0();
    __syncthreads();

    const int n0 = blockIdx.x * 64 + wave * 16;
    if (n0 >= N) return;

    const int h  = lane >> 4;
    const int mn = lane & 15;
    const float* Brow = Bm + (size_t)(n0 + mn) * D_DIM;
    const float* Alds = &ldsA[mn * LDSA_STRIDE];

    v8f acc = {0.f, 0.f, 0.f, 0.f, 0.f, 0.f, 0.f, 0.f};
#pragma unroll 4
    for (int k0 = 0; k0 < D_DIM; k0 += 4) {
        if ((k0 & 63) == 0)
            __builtin_prefetch((const void*)(Brow + k0 + 64), 0, 0);
        v2f a, bf;
        a.x  = Alds[k0 + 2 * h];
        a.y  = Alds[k0 + 2 * h + 1];
        bf.x = Brow[k0 + 2 * h];
        bf.y = Brow[k0 + 2 * h + 1];
        acc = __builtin_amdgcn_wmma_f32_16x16x4_f32(
            false, a, false, bf, (short)0, acc, false, false);
    }

    const float bval = bias ? bias[n0 + mn] : 0.0f;
#pragma unroll
    for (int r = 0; r < 8; ++r) {
        const int row = r0 + r + 8 * h;
        C[(size_t)row * (size_t)N + (n0 + mn)] = acc[r] + bval;
    }
}

// ---------------------------------------------------------------------------
// Kernel 5: masked 16-way cross-attention.  One block per batch row.
// ---------------------------------------------------------------------------
__global__ __launch_bounds__(256)
void attn_kernel(const float* __restrict__ Qp, const float* __restrict__ Kp,
                 const float* __restrict__ Vp, const float* __restrict__ topVal,
                 float* __restrict__ memAtt, float* __restrict__ maxSim) {
    const int b = blockIdx.x;
    const int t = threadIdx.x;
    __shared__ float red[256];
    __shared__ float sc[TOPK];
    __shared__ float attn[TOPK];

    const float* qrow = Qp + (size_t)b * D_DIM;
    const float q0 = qrow[t], q1 = qrow[t + 256];

    for (int k = 0; k < TOPK; ++k) {
        const float* krow = Kp + ((size_t)b * TOPK + k) * D_DIM;
        const float p = q0 * krow[t] + q1 * krow[t + 256];
        const float s = blockReduceSum256(p, red);
        if (t == 0) sc[k] = s * SCALE_F;
    }
    __syncthreads();

    if (t == 0) {
        float tv[TOPK], s[TOPK];
        float msim = neg_inf();
        for (int k = 0; k < TOPK; ++k) { tv[k] = topVal[b * TOPK + k]; msim = fmaxf(msim, tv[k]); }
        maxSim[b] = msim;
        float mx = neg_inf();
        for (int k = 0; k < TOPK; ++k) {
            s[k] = (tv[k] > neg_inf()) ? sc[k] : neg_inf();
            mx = fmaxf(mx, s[k]);
        }
        float sum = 0.f;
        for (int k = 0; k < TOPK; ++k) {
            const float e = (s[k] > neg_inf()) ? __expf(s[k] - mx) : 0.f;
            attn[k] = e; sum += e;
        }
        const float inv = (sum > 0.f) ? 1.f / sum : 0.f;
        for (int k = 0; k < TOPK; ++k) attn[k] *= inv;
    }
    __syncthreads();

    float o0 = 0.f, o1 = 0.f;
    for (int k = 0; k < TOPK; ++k) {
        const float* vrow = Vp + ((size_t)b * TOPK + k) * D_DIM;
        o0 += attn[k] * vrow[t];
        o1 += attn[k] * vrow[t + 256];
    }
    memAtt[b * D_DIM + t]       = o0;
    memAtt[b * D_DIM + t + 256] = o1;
}

// ---------------------------------------------------------------------------
// Kernel 6: gate * conf fusion + LayerNorm + delta.  One block per batch row.
// ---------------------------------------------------------------------------
__global__ __launch_bounds__(256)
void gate_ln_kernel(const float* __restrict__ series, const float* __restrict__ memOut,
                    const float* __restrict__ Wg, const float* __restrict__ bg,
                    const float* __restrict__ maxSim, const int* __restrict__ anyFlag,
                    const float* __restrict__ lng, const float* __restrict__ lnb,
                    float* __restrict__ delta) {
    const int b = blockIdx.x;
    const int t = threadIdx.x;
    __shared__ float red[256];
    const float* srow = series + (size_t)b * D_DIM;
    const float* mrow = memOut + (size_t)b * D_DIM;
    const float s0 = srow[t], s1 = srow[t + 256];
    const float m0 = mrow[t], m1 = mrow[t + 256];

    const float dot = blockReduceSum256(
        s0 * Wg[t] + s1 * Wg[t + 256] + m0 * Wg[512 + t] + m1 * Wg[768 + t], red);
    const float g  = 1.f / (1.f + __expf(-(dot + bg[0])));
    const float cf = 1.f / (1.f + __expf(-maxSim[b]));    // sigmoid(-inf) -> 0
    const float gc = g * cf;

    const float f0 = s0 + gc * m0;
    const float f1 = s1 + gc * m1;
    const float mu = blockReduceSum256(f0 + f1, red) * (1.0f / (float)D_DIM);
    const float va = blockReduceSum256((f0 - mu) * (f0 - mu) + (f1 - mu) * (f1 - mu), red)
                     * (1.0f / (float)D_DIM);
    const float rstd = rsqrtf(va + LN_EPS);
    float n0 = (f0 - mu) * rstd * lng[t]       + lnb[t];
    float n1 = (f1 - mu) * rstd * lng[t + 256] + lnb[t + 256];
    if (*anyFlag == 0) { n0 = s0; n1 = s1; }
    delta[b * D_DIM + t]       = n0 - s0;
    delta[b * D_DIM + t + 256] = n1 - s1;
}

// ---------------------------------------------------------------------------
// Kernel 7: out = hidden + delta broadcast over S.  float4 streaming.
// ---------------------------------------------------------------------------
__global__ __launch_bounds__(256)
void broadcast_add_kernel(const float* __restrict__ hidden,
                          const float* __restrict__ delta,
                          float* __restrict__ out) {
    const size_t v = (size_t)blockIdx.x * 256 + threadIdx.x;
    const size_t e = v * 4;
    const int b = (int)(e >> 18);                 // / (S*D)
    const int d = (int)(e & (D_DIM - 1));
    const float4 h  = *(const float4*)(hidden + e);
    const float4 dd = *(const float4*)(delta + (size_t)b * D_DIM + d);
    float4 o;
    o.x = h.x + dd.x; o.y = h.y + dd.y; o.z = h.z + dd.z; o.w = h.w + dd.w;
    *(float4*)(out + e) = o;
}

// ---------------------------------------------------------------------------
// Launch
// ---------------------------------------------------------------------------
extern "C" void kernel_launch(void* const* d_in, const int* in_sizes, int n_in,
                              void* d_out, int out_size, void* d_ws, size_t ws_size,
                              hipStream_t stream) {
    (void)in_sizes; (void)n_in; (void)out_size; (void)ws_size;
    const float* hidden = (const float*)d_in[0];
    const float* bank   = (const float*)d_in[1];
    const float* Wq = (const float*)d_in[2];  const float* bq = (const float*)d_in[3];
    const float* Wk = (const float*)d_in[4];  const float* bk = (const float*)d_in[5];
    const float* Wv = (const float*)d_in[6];  const float* bv = (const float*)d_in[7];
    const float* Wo = (const float*)d_in[8];  const float* bo = (const float*)d_in[9];
    const float* Wg = (const float*)d_in[10]; const float* bg = (const float*)d_in[11];
    const float* lng = (const float*)d_in[12]; const float* lnb = (const float*)d_in[13];
    float* out = (float*)d_out;

    // workspace layout (float offsets); total ~37.5 MB
    float* ws      = (float*)d_ws;
    float* series  = ws + 0;          // 256*512
    float* q       = ws + 131072;     // 256*512
    float* Qp      = ws + 262144;     // 256*512
    float* memAtt  = ws + 393216;     // 256*512
    float* memOut  = ws + 524288;     // 256*512
    float* delta   = ws + 655360;     // 256*512
    float* maxSim  = ws + 786432;     // 256
    float* topVal  = ws + 786688;     // 256*16
    int*   topIdx  = (int*)(ws + 790784);   // 256*16
    int*   anyFlag = (int*)(ws + 794880);   // 1
    float* Kp      = ws + 1048576;    // 4096*512
    float* Vp      = ws + 3145728;    // 4096*512
    float* candVal = ws + 5242880;    // 256*63*128 = 2064384
    int*   candIdx = (int*)(ws + 7307264);  // 2064384

    // 1) pool + normalize (resets anyFlag)
    pool_kernel<<<256, 256, 0, stream>>>(hidden, series, q, anyFlag);
    // 2) fused sims GEMM + partial top-16 (row-blocks fastest for L2 reuse)
    sims_topk_partial<<<dim3(16, GX_SIMS), 128, 0, stream>>>(q, bank, candVal, candIdx);
    // 3) exact top-16 merge
    topk_merge<<<256, 256, 0, stream>>>(candVal, candIdx, topVal, topIdx, anyFlag);
    // 4) Qp = series @ Wq^T + bq
    wmma_gemm_nt<<<dim3(8, 16), 128, 0, stream>>>(series, Wq, nullptr, bq, Qp, D_DIM);
    // 5,6) Kp/Vp = gather(bank, topIdx) @ {Wk,Wv}^T + b
    wmma_gemm_nt<<<dim3(8, 256), 128, 0, stream>>>(bank, Wk, topIdx, bk, Kp, D_DIM);
    wmma_gemm_nt<<<dim3(8, 256), 128, 0, stream>>>(bank, Wv, topIdx, bv, Vp, D_DIM);
    // 7) masked softmax attention over the 16 retrieved memories
    attn_kernel<<<256, 256, 0, stream>>>(Qp, Kp, Vp, topVal, memAtt, maxSim);
    // 8) mem_out = memAtt @ Wo^T + bo
    wmma_gemm_nt<<<dim3(8, 16), 128, 0, stream>>>(memAtt, Wo, nullptr, bo, memOut, D_DIM);
    // 9) gate * conf + LayerNorm -> delta
    gate_ln_kernel<<<256, 256, 0, stream>>>(series, memOut, Wg, bg, maxSim, anyFlag,
                                            lng, lnb, delta);
    // 10) out = hidden + delta (broadcast over S)
    broadcast_add_kernel<<<65536, 256, 0, stream>>>(hidden, delta, out);
}